// SaccadicViTEmbeddings_55061480735263
// MI455X (gfx1250) — compile-verified
//
#include <hip/hip_runtime.h>
#include <hip/hip_bf16.h>

typedef __attribute__((ext_vector_type(2))) float v2f;
typedef __attribute__((ext_vector_type(8))) float v8f;

#define CUBIC_A (-0.75f)

__device__ __forceinline__ void cubic_weights(float t, float w[4]) {
    const float t1 = t + 1.0f;
    const float t4 = 2.0f - t;
    w[0] = ((CUBIC_A * t1 - 5.0f * CUBIC_A) * t1 + 8.0f * CUBIC_A) * t1 - 4.0f * CUBIC_A;
    w[1] = ((CUBIC_A + 2.0f) * t - (CUBIC_A + 3.0f)) * t * t + 1.0f;
    const float s = 1.0f - t;
    w[2] = ((CUBIC_A + 2.0f) * s - (CUBIC_A + 3.0f)) * s * s + 1.0f;
    w[3] = ((CUBIC_A * t4 - 5.0f * CUBIC_A) * t4 + 8.0f * CUBIC_A) * t4 - 4.0f * CUBIC_A;
}

// -----------------------------------------------------------------------------
// Kernel 1: bicubic patch sampling.
// One block per batch image (512 blocks, 256 threads = 8 wave32).
// Threads 0..31 precompute the separable x/y weights (masked) and clipped
// indices into LDS; then each thread owns one (p,q) grid position and
// accumulates the 4x4 taps for all 3 channels.
// Output layout: patches[b][k], k = c*256 + p*16 + q  (row-major [512][768]),
// i.e. exactly the A matrix of the downstream GEMM.
// -----------------------------------------------------------------------------
__global__ void __launch_bounds__(256)
saccade_sample_kernel(const float* __restrict__ px,   // (512,3,224,224)
                      const float* __restrict__ pc,   // (512,2)
                      float* __restrict__ patches)    // (512,768) ws
{
    const int b   = blockIdx.x;
    const int tid = threadIdx.x;

    __shared__ float s_w[2][16][4];   // [0]=x weights*mask, [1]=y
    __shared__ int   s_i[2][16][4];   // clipped indices

    if (tid < 32) {
        const int axis = tid >> 4;            // 0 = x, 1 = y
        const int p    = tid & 15;
        const float t     = pc[b * 2 + axis];
        const float lin   = -1.0f + (2.0f / 15.0f) * (float)p;
        const float coord = ((lin + t + 1.0f) * 224.0f - 1.0f) * 0.5f;
        const float f0    = floorf(coord);
        float w[4];
        cubic_weights(coord - f0, w);
        const int base = (int)f0;
        #pragma unroll
        for (int j = 0; j < 4; ++j) {
            const int idx = base + j - 1;
            const float m = (idx >= 0 && idx < 224) ? 1.0f : 0.0f;
            s_w[axis][p][j] = w[j] * m;
            s_i[axis][p][j] = min(max(idx, 0), 223);
        }
    }
    __syncthreads();

    const int p = tid >> 4;     // row of grid (y)
    const int q = tid & 15;     // col of grid (x)

    float wx[4]; int xi[4];
    float wy[4]; int yi[4];
    #pragma unroll
    for (int j = 0; j < 4; ++j) {
        wx[j] = s_w[0][q][j]; xi[j] = s_i[0][q][j];
        wy[j] = s_w[1][p][j]; yi[j] = s_i[1][p][j];
    }

    #pragma unroll
    for (int c = 0; c < 3; ++c) {
        const float* img = px + ((size_t)(b * 3 + c)) * 224 * 224;
        float acc = 0.0f;
        #pragma unroll
        for (int i = 0; i < 4; ++i) {
            const float* row = img + yi[i] * 224;
            float r = 0.0f;
            #pragma unroll
            for (int j = 0; j < 4; ++j)
                r = fmaf(wx[j], row[xi[j]], r);
            acc = fmaf(wy[i], r, acc);
        }
        patches[(size_t)b * 768 + c * 256 + tid] = acc;
    }
}

// -----------------------------------------------------------------------------
// Kernel 2: FP32 WMMA GEMM (M=512, N=768, K=768) + fused epilogue.
//   out[m][n] = sum_k patches[m][k] * conv_w[n][k]
//             + conv_b[n] + pc[m][0]*pe_w[n][0] + pc[m][1]*pe_w[n][1] + pe_b[n]
// One wave32 per 16x16 tile; grid = (48 N-tiles, 32 M-tiles).
// V_WMMA_F32_16X16X4_F32 layouts (ISA 7.12.2):
//   A 16x4 : lane l -> A[l&15][(l>>4)*2 + {0,1}]        (contiguous 8B load)
//   B 4x16 : lane l -> B[(l>>4)*2 + {0,1}][l&15]
//            = conv_w[n_base+(l&15)][k + (l>>4)*2 + {0,1}] (contiguous 8B load)
//   C/D    : VGPR r -> row r + 8*(l>>4), col l&15
// -----------------------------------------------------------------------------
__global__ void __launch_bounds__(32)
saccade_gemm_wmma_kernel(const float* __restrict__ Apat,  // (512,768) ws
                         const float* __restrict__ Wc,    // conv_w (768,768)
                         const float* __restrict__ cb,    // conv_b (768)
                         const float* __restrict__ pc,    // (512,2)
                         const float* __restrict__ pew,   // (768,2)
                         const float* __restrict__ peb,   // (768)
                         float* __restrict__ out)         // (512,768)
{
    const int n_base = blockIdx.x * 16;
    const int m_base = blockIdx.y * 16;
    const int lane   = threadIdx.x;
    const int half   = lane >> 4;   // which K-half of the 4-wide slab
    const int l15    = lane & 15;

    const float* arow = Apat + (size_t)(m_base + l15) * 768 + half * 2;
    const float* brow = Wc   + (size_t)(n_base + l15) * 768 + half * 2;

    v8f c = {};
    #pragma unroll 8
    for (int k = 0; k < 768; k += 4) {
        const v2f a = *(const v2f*)(arow + k);
        const v2f b = *(const v2f*)(brow + k);
        // 8 args: (neg_a, A, neg_b, B, c_mod, C, reuse_a, reuse_b)
        c = __builtin_amdgcn_wmma_f32_16x16x4_f32(false, a, false, b,
                                                  (short)0, c, false, false);
    }

    // Epilogue: per-lane column n is fixed; rows vary with VGPR index r.
    const int n = n_base + l15;
    const float addn = cb[n] + peb[n];
    const float pw0  = pew[n * 2 + 0];
    const float pw1  = pew[n * 2 + 1];

    #pragma unroll
    for (int r = 0; r < 8; ++r) {
        const int m = m_base + r + half * 8;
        const float pos = fmaf(pc[m * 2 + 0], pw0, pc[m * 2 + 1] * pw1);
        out[(size_t)m * 768 + n] = c[r] + addn + pos;
    }
}

extern "C" void kernel_launch(void* const* d_in, const int* in_sizes, int n_in,
                              void* d_out, int out_size, void* d_ws, size_t ws_size,
                              hipStream_t stream) {
    const float* pixel_values = (const float*)d_in[0];  // (512,3,224,224)
    const float* patch_config = (const float*)d_in[1];  // (512,2)
    const float* conv_w       = (const float*)d_in[2];  // (768,3,16,16)
    const float* conv_b       = (const float*)d_in[3];  // (768)
    const float* pe_w         = (const float*)d_in[4];  // (768,2)
    const float* pe_b         = (const float*)d_in[5];  // (768)
    float*       out          = (float*)d_out;          // (512,768)

    float* patches = (float*)d_ws;                      // 512*768*4 = 1.5 MB

    // Phase 1: bicubic sampling -> patches (A matrix of the GEMM)
    saccade_sample_kernel<<<512, 256, 0, stream>>>(pixel_values, patch_config, patches);

    // Phase 2: FP32 WMMA GEMM + fused bias/positional epilogue
    dim3 grid(48, 32);   // N tiles x M tiles
    saccade_gemm_wmma_kernel<<<grid, 32, 0, stream>>>(patches, conv_w, conv_b,
                                                      patch_config, pe_w, pe_b, out);
}